// PageRank_63934883169038
// MI455X (gfx1250) — compile-verified
//
#include <hip/hip_runtime.h>
#include <hip/hip_bf16.h>
#include <stdint.h>

// ---------------------------------------------------------------------------
// GCNConv(1,1)-style message passing, factored passes:
//   deg[c]  = #edges into c                          (u32 scatter-add, L2)
//   dis[i]  = deg>0 ? rsqrt(deg) : 0                 (node pass, in-place)
//   sx[i]   = dis[i] * x[i]
//   t[c]    = sum_{e: col=c} sx[row[e]]              (f32 scatter-add, L2)
//   xnew[i] = sigmoid(x*nw + dis[i]*t[i] + bias)
//   out[p]  = sign(xnew[home]-xnew[away])
//
// MI455X specifics:
//  * Edge-index tiles staged into LDS with gfx1250 async DMA
//    (global_load_async_to_lds_b128, ASYNCcnt), double-buffered:
//    issue tile i+1, s_wait_asynccnt 2, consume tile i. In-order async
//    completion makes the "<=2" wait release exactly the previous tile.
//  * Edge stream is 128MB read-once -> th:TH_LOAD_NT so it does not evict
//    the ~10MB atomic-resident node arrays from the 192MB L2.
//  * Each lane consumes only LDS bytes its own async op wrote -> no barriers.
// ---------------------------------------------------------------------------

#define THREADS 256
#define EPT 4                    // edges per thread per tile (b128 = 4 x i32)
#define TILE (THREADS * EPT)     // 1024 edges per tile
#define ITERS 4                  // tiles per block
#define BLOCK_EDGES (TILE * ITERS)

__device__ __forceinline__ unsigned lds_off_of(const void* p) {
  // low 32 bits of a shared-aperture pointer = wave-relative LDS byte offset
  return (unsigned)(uintptr_t)p;
}
__device__ __forceinline__ void async_copy_b128_nt(unsigned lds_off, const void* gaddr) {
  asm volatile("global_load_async_to_lds_b128 %0, %1, off th:TH_LOAD_NT"
               :: "v"(lds_off), "v"((unsigned long long)(uintptr_t)gaddr)
               : "memory");
}
__device__ __forceinline__ void wait_async_le(int n) {
  if (n == 0) asm volatile("s_wait_asynccnt 0x0" ::: "memory");
  else if (n == 1) asm volatile("s_wait_asynccnt 0x1" ::: "memory");
  else asm volatile("s_wait_asynccnt 0x2" ::: "memory");
}

// ---- pass 0: zero workspace (ws is not re-poisoned between replays) --------
__global__ void init_ws_kernel(unsigned* __restrict__ deg, float* __restrict__ t, int N) {
  int i = blockIdx.x * blockDim.x + threadIdx.x;
  if (i < N) { deg[i] = 0u; t[i] = 0.0f; }
}

// ---- pass 1: in-degree at target (col), double-buffered async staging ------
__global__ void deg_pass_kernel(const int* __restrict__ col, unsigned* __restrict__ deg,
                                int E) {
  __shared__ __align__(16) int s_col[2][TILE];
  const int tid = threadIdx.x;
  const int lane4 = tid * EPT;
  const long long block_base = (long long)blockIdx.x * BLOCK_EDGES;
  const bool fast = (((uintptr_t)col & 15u) == 0u) &&
                    (block_base + BLOCK_EDGES <= (long long)E);

  if (fast) {
    async_copy_b128_nt(lds_off_of(&s_col[0][lane4]), col + block_base + lane4);
    #pragma unroll
    for (int it = 0; it < ITERS; ++it) {
      const int cur = it & 1, nxt = cur ^ 1;
      if (it + 1 < ITERS) {
        async_copy_b128_nt(lds_off_of(&s_col[nxt][lane4]),
                           col + block_base + (long long)(it + 1) * TILE + lane4);
        wait_async_le(1);        // the just-issued op may be outstanding
      } else {
        wait_async_le(0);
      }
      #pragma unroll
      for (int k = 0; k < EPT; ++k) {
        atomicAdd(&deg[s_col[cur][lane4 + k]], 1u);
      }
    }
  } else {
    for (int it = 0; it < ITERS; ++it) {
      long long tbase = block_base + (long long)it * TILE + lane4;
      #pragma unroll
      for (int k = 0; k < EPT; ++k) {
        long long e = tbase + k;
        if (e < (long long)E) {
          int c = __builtin_nontemporal_load(col + e);
          atomicAdd(&deg[c], 1u);
        }
      }
    }
  }
}

// ---- pass 2: dis = rsqrt(deg) (in-place over deg), sx = dis * x ------------
__global__ void node_pass_kernel(const float* __restrict__ x, unsigned* deg_dis,
                                 float* __restrict__ sx, int N) {
  int i = blockIdx.x * blockDim.x + threadIdx.x;
  if (i < N) {
    unsigned d = deg_dis[i];
    float dis = (d != 0u) ? rsqrtf((float)d) : 0.0f;
    reinterpret_cast<float*>(deg_dis)[i] = dis;  // overwrite u32 deg with f32 dis
    sx[i] = dis * x[i];
  }
}

// ---- pass 3: t[col] += sx[row]  (dis[col] factored out, applied later) -----
__global__ void msg_pass_kernel(const int* __restrict__ row, const int* __restrict__ col,
                                const float* __restrict__ sx, float* __restrict__ t,
                                int E) {
  __shared__ __align__(16) int s_row[2][TILE];
  __shared__ __align__(16) int s_col[2][TILE];
  const int tid = threadIdx.x;
  const int lane4 = tid * EPT;
  const long long block_base = (long long)blockIdx.x * BLOCK_EDGES;
  const bool fast = (((((uintptr_t)row) | ((uintptr_t)col)) & 15u) == 0u) &&
                    (block_base + BLOCK_EDGES <= (long long)E);

  if (fast) {
    async_copy_b128_nt(lds_off_of(&s_row[0][lane4]), row + block_base + lane4);
    async_copy_b128_nt(lds_off_of(&s_col[0][lane4]), col + block_base + lane4);
    #pragma unroll
    for (int it = 0; it < ITERS; ++it) {
      const int cur = it & 1, nxt = cur ^ 1;
      if (it + 1 < ITERS) {
        const long long nbase = block_base + (long long)(it + 1) * TILE + lane4;
        async_copy_b128_nt(lds_off_of(&s_row[nxt][lane4]), row + nbase);
        async_copy_b128_nt(lds_off_of(&s_col[nxt][lane4]), col + nbase);
        wait_async_le(2);        // 2 just-issued ops outstanding; tile `cur` landed
      } else {
        wait_async_le(0);
      }
      #pragma unroll
      for (int k = 0; k < EPT; ++k) {
        int r = s_row[cur][lane4 + k];
        int c = s_col[cur][lane4 + k];
        atomicAdd(&t[c], sx[r]);   // gather + scatter both L2-resident (2MB each)
      }
    }
  } else {
    for (int it = 0; it < ITERS; ++it) {
      long long tbase = block_base + (long long)it * TILE + lane4;
      #pragma unroll
      for (int k = 0; k < EPT; ++k) {
        long long e = tbase + k;
        if (e < (long long)E) {
          int r = __builtin_nontemporal_load(row + e);
          int c = __builtin_nontemporal_load(col + e);
          atomicAdd(&t[c], sx[r]);
        }
      }
    }
  }
}

// ---- pass 4: pairs: sign(sigmoid(z_h) - sigmoid(z_a)) ----------------------
__global__ void out_pass_kernel(const float* __restrict__ x, const float* __restrict__ nw,
                                const float* __restrict__ bias,
                                const int* __restrict__ home, const int* __restrict__ away,
                                const float* __restrict__ dis, const float* __restrict__ t,
                                float* __restrict__ out, int P) {
  int p = blockIdx.x * blockDim.x + threadIdx.x;
  if (p < P) {
    const float b = bias[0];
    int h = home[p], a = away[p];
    float zh = fmaf(x[h], nw[h], fmaf(dis[h], t[h], b));
    float za = fmaf(x[a], nw[a], fmaf(dis[a], t[a], b));
    float sh = 1.0f / (1.0f + __expf(-zh));
    float sa = 1.0f / (1.0f + __expf(-za));
    float d = sh - sa;
    out[p] = (d > 0.0f) ? 1.0f : ((d < 0.0f) ? -1.0f : 0.0f);
  }
}

extern "C" void kernel_launch(void* const* d_in, const int* in_sizes, int n_in,
                              void* d_out, int out_size, void* d_ws, size_t ws_size,
                              hipStream_t stream) {
  const float* x    = (const float*)d_in[0];   // [N]
  const float* nw   = (const float*)d_in[1];   // [N]
  const float* bias = (const float*)d_in[2];   // [1]
  const int*   edge = (const int*)d_in[3];     // [2*E]: row half then col half
  const int*   home = (const int*)d_in[4];     // [P]
  const int*   away = (const int*)d_in[5];     // [P]

  const int N = in_sizes[0];
  const int E = in_sizes[3] / 2;
  const int P = in_sizes[4];

  const int* row = edge;
  const int* col = edge + E;

  // Workspace layout: [deg(u32)->dis(f32) : N][sx(f32) : N][t(f32) : N]
  char* ws = (char*)d_ws;
  unsigned* deg = (unsigned*)ws;
  float*    sx  = (float*)(ws + (size_t)N * 4u);
  float*    t   = (float*)(ws + (size_t)N * 8u);
  float*    out = (float*)d_out;

  const int nb_node = (N + THREADS - 1) / THREADS;
  const int nb_edge = (int)(((long long)E + BLOCK_EDGES - 1) / BLOCK_EDGES);
  const int nb_pair = (P + THREADS - 1) / THREADS;

  init_ws_kernel<<<nb_node, THREADS, 0, stream>>>(deg, t, N);
  deg_pass_kernel<<<nb_edge, THREADS, 0, stream>>>(col, deg, E);
  node_pass_kernel<<<nb_node, THREADS, 0, stream>>>(x, deg, sx, N);
  msg_pass_kernel<<<nb_edge, THREADS, 0, stream>>>(row, col, sx, t, E);
  out_pass_kernel<<<nb_pair, THREADS, 0, stream>>>(x, nw, bias, home, away,
                                                   (const float*)deg, t, out, P);
}